// MPN_57243324121109
// MI455X (gfx1250) — compile-verified
//
#include <hip/hip_runtime.h>

#define NATOMS 100000
#define NBONDS 200000
#define NMOLS  2000
#define AFD    18
#define BFD    5
#define BIN    23      // AFD + BFD
#define MAXNB  15
#define H      128
#define DEPTH  6

typedef __attribute__((ext_vector_type(2))) float v2f;
typedef __attribute__((ext_vector_type(8))) float v8f;

__device__ __forceinline__ v8f wmma_f32(v2f a, v2f b, v8f c) {
  // D = A(16x4) * B(4x16) + C(16x16), fp32 throughout -> v_wmma_f32_16x16x4_f32
  return __builtin_amdgcn_wmma_f32_16x16x4_f32(false, a, false, b, (short)0, c,
                                               false, false);
}

// ---------------------------------------------------------------------------
// Kernel 1: binput = fbonds @ W_i ; message = relu(binput)
// Block = 256 threads, 32 bonds. K=23 zero-padded to 24.
// ---------------------------------------------------------------------------
__global__ __launch_bounds__(256) void k_bond_input(
    const float* __restrict__ fbonds, const float* __restrict__ Wi,
    float* __restrict__ binput, float* __restrict__ msg) {
  __shared__ float tA[32][26];            // stride 26: conflict-free column reads
  const int e0  = blockIdx.x * 32;
  const int tid = threadIdx.x;
  for (int i = tid; i < 32 * 24; i += 256) {
    int r = i / 24, c = i % 24;
    tA[r][c] = (c < BIN) ? fbonds[(size_t)(e0 + r) * BIN + c] : 0.0f;
  }
  __syncthreads();
  const int wave = tid >> 5;              // N tile (0..7)
  const int lane = tid & 31;
  const int m    = lane & 15;
  const int kh   = (lane < 16) ? 0 : 2;
  const int col  = wave * 16 + (lane & 15);
  v8f acc0 = {}; v8f acc1 = {};
#pragma unroll
  for (int kk = 0; kk < 24; kk += 4) {
    int k = kk + kh;
    v2f b, a0, a1;
    b.x  = (k     < BIN) ? Wi[(size_t)k       * H + col] : 0.0f;
    b.y  = (k + 1 < BIN) ? Wi[(size_t)(k + 1) * H + col] : 0.0f;
    a0.x = tA[m][k];      a0.y = tA[m][k + 1];
    a1.x = tA[m + 16][k]; a1.y = tA[m + 16][k + 1];
    acc0 = wmma_f32(a0, b, acc0);
    acc1 = wmma_f32(a1, b, acc1);
  }
  const int rhalf = (lane < 16) ? 0 : 8;
#pragma unroll
  for (int g = 0; g < 8; ++g) {
    int r = g + rhalf;
    size_t o0 = (size_t)(e0 + r)      * H + col;
    size_t o1 = (size_t)(e0 + 16 + r) * H + col;
    float v0 = acc0[g], v1 = acc1[g];
    binput[o0] = v0; msg[o0] = fmaxf(v0, 0.0f);
    binput[o1] = v1; msg[o1] = fmaxf(v1, 0.0f);
  }
}

// ---------------------------------------------------------------------------
// Kernel 2 (x5): msg_out = relu(binput + gathersum(msg_in, bgraph) @ W_h)
// Fused: gather-sum into LDS (L2-resident message), then WMMA GEMM.
// ---------------------------------------------------------------------------
__global__ __launch_bounds__(256) void k_depth(
    const float* __restrict__ msg_in, const float* __restrict__ binput,
    const int* __restrict__ bgraph, const float* __restrict__ Wh,
    float* __restrict__ msg_out) {
  __shared__ float nei[32][132];          // 132 = 128 + pad -> 16 distinct banks
  const int e0  = blockIdx.x * 32;
  const int tid = threadIdx.x;
  {                                       // phase 1: gather-sum (8 thr / bond)
    const int r = tid >> 3;
    const int p = tid & 7;                // 16-float chunk
    float acc[16];
#pragma unroll
    for (int j = 0; j < 16; ++j) acc[j] = 0.0f;
    const int* gr = bgraph + (size_t)(e0 + r) * MAXNB;
    for (int nb = 0; nb < MAXNB; ++nb) {
      int idx = gr[nb];
      const float4* src =
          reinterpret_cast<const float4*>(msg_in + (size_t)idx * H + p * 16);
#pragma unroll
      for (int j = 0; j < 4; ++j) {
        float4 v = src[j];
        acc[4 * j] += v.x; acc[4 * j + 1] += v.y;
        acc[4 * j + 2] += v.z; acc[4 * j + 3] += v.w;
      }
    }
#pragma unroll
    for (int j = 0; j < 16; ++j) nei[r][p * 16 + j] = acc[j];
  }
  __syncthreads();
  const int wave = tid >> 5;
  const int lane = tid & 31;
  const int m    = lane & 15;
  const int kh   = (lane < 16) ? 0 : 2;
  const int col  = wave * 16 + (lane & 15);
  v8f acc0 = {}; v8f acc1 = {};
#pragma unroll
  for (int kk = 0; kk < H; kk += 4) {     // 32 K-steps, 64 WMMA per wave
    int k = kk + kh;
    v2f b, a0, a1;
    b.x  = Wh[(size_t)k       * H + col];
    b.y  = Wh[(size_t)(k + 1) * H + col];
    a0.x = nei[m][k];      a0.y = nei[m][k + 1];
    a1.x = nei[m + 16][k]; a1.y = nei[m + 16][k + 1];
    acc0 = wmma_f32(a0, b, acc0);
    acc1 = wmma_f32(a1, b, acc1);
  }
  const int rhalf = (lane < 16) ? 0 : 8;
#pragma unroll
  for (int g = 0; g < 8; ++g) {
    int r = g + rhalf;
    size_t o0 = (size_t)(e0 + r)      * H + col;
    size_t o1 = (size_t)(e0 + 16 + r) * H + col;
    msg_out[o0] = fmaxf(binput[o0] + acc0[g], 0.0f);
    msg_out[o1] = fmaxf(binput[o1] + acc1[g], 0.0f);
  }
}

// ---------------------------------------------------------------------------
// Kernel 3: atom readout. ainput = [fatoms | gathersum(msg, agraph)] (K=146),
// atomh = relu(ainput @ W_o + b_o). K padded to 148.
// ---------------------------------------------------------------------------
__global__ __launch_bounds__(256) void k_atom(
    const float* __restrict__ msg, const float* __restrict__ fatoms,
    const int* __restrict__ agraph, const float* __restrict__ Wo,
    const float* __restrict__ bo, float* __restrict__ atomh) {
  __shared__ float ain[32][148];          // stride 148: conflict-free columns
  const int a0  = blockIdx.x * 32;
  const int tid = threadIdx.x;
  for (int i = tid; i < 32 * AFD; i += 256) {
    int r = i / AFD, c = i % AFD;
    ain[r][c] = fatoms[(size_t)(a0 + r) * AFD + c];
  }
  if (tid < 64) {                         // zero pad cols 146..147
    ain[tid >> 1][146 + (tid & 1)] = 0.0f;
  }
  {
    const int r = tid >> 3;
    const int p = tid & 7;
    float acc[16];
#pragma unroll
    for (int j = 0; j < 16; ++j) acc[j] = 0.0f;
    const int* gr = agraph + (size_t)(a0 + r) * MAXNB;
    for (int nb = 0; nb < MAXNB; ++nb) {
      int idx = gr[nb];
      const float4* src =
          reinterpret_cast<const float4*>(msg + (size_t)idx * H + p * 16);
#pragma unroll
      for (int j = 0; j < 4; ++j) {
        float4 v = src[j];
        acc[4 * j] += v.x; acc[4 * j + 1] += v.y;
        acc[4 * j + 2] += v.z; acc[4 * j + 3] += v.w;
      }
    }
#pragma unroll
    for (int j = 0; j < 16; ++j) ain[r][AFD + p * 16 + j] = acc[j];
  }
  __syncthreads();
  const int wave = tid >> 5;
  const int lane = tid & 31;
  const int m    = lane & 15;
  const int kh   = (lane < 16) ? 0 : 2;
  const int col  = wave * 16 + (lane & 15);
  v8f acc0 = {}; v8f acc1 = {};
#pragma unroll
  for (int kk = 0; kk < 148; kk += 4) {   // 37 K-steps
    int k = kk + kh;
    v2f b, a0f, a1f;
    b.x   = (k     < 146) ? Wo[(size_t)k       * H + col] : 0.0f;
    b.y   = (k + 1 < 146) ? Wo[(size_t)(k + 1) * H + col] : 0.0f;
    a0f.x = ain[m][k];      a0f.y = ain[m][k + 1];
    a1f.x = ain[m + 16][k]; a1f.y = ain[m + 16][k + 1];
    acc0 = wmma_f32(a0f, b, acc0);
    acc1 = wmma_f32(a1f, b, acc1);
  }
  const float bias = bo[col];
  const int rhalf = (lane < 16) ? 0 : 8;
#pragma unroll
  for (int g = 0; g < 8; ++g) {
    int r = g + rhalf;
    atomh[(size_t)(a0 + r)      * H + col] = fmaxf(acc0[g] + bias, 0.0f);
    atomh[(size_t)(a0 + 16 + r) * H + col] = fmaxf(acc1[g] + bias, 0.0f);
  }
}

// ---------------------------------------------------------------------------
// Kernel 4: ragged segment mean pooling. One block per molecule.
// ---------------------------------------------------------------------------
__global__ __launch_bounds__(128) void k_pool(
    const float* __restrict__ atomh, const int* __restrict__ start,
    const int* __restrict__ len, float* __restrict__ out) {
  const int mol = blockIdx.x;
  const int h   = threadIdx.x;
  const int s   = start[mol];
  const int L   = len[mol];
  float sum = 0.0f;
  for (int i = 0; i < L; ++i) sum += atomh[(size_t)(s + i) * H + h];
  out[(size_t)mol * H + h] = sum / (float)L;
}

// ---------------------------------------------------------------------------
extern "C" void kernel_launch(void* const* d_in, const int* in_sizes, int n_in,
                              void* d_out, int out_size, void* d_ws,
                              size_t ws_size, hipStream_t stream) {
  const float* fatoms      = (const float*)d_in[0];
  const float* fbonds      = (const float*)d_in[1];
  const int*   agraph      = (const int*)d_in[2];
  const int*   bgraph      = (const int*)d_in[3];
  const int*   scope_start = (const int*)d_in[4];
  const int*   scope_len   = (const int*)d_in[5];
  const float* Wi          = (const float*)d_in[6];
  const float* Wh          = (const float*)d_in[7];
  const float* Wo          = (const float*)d_in[8];
  const float* bo          = (const float*)d_in[9];
  float* out = (float*)d_out;

  // workspace: binput[E,H] | msgA[E,H] | msgB[E,H] | atomh[N,H]  (~358 MB)
  float* ws = (float*)d_ws;
  const size_t EH = (size_t)NBONDS * H;
  float* binput = ws;
  float* msgA   = ws + EH;
  float* msgB   = ws + 2 * EH;
  float* atomh  = ws + 3 * EH;

  k_bond_input<<<NBONDS / 32, 256, 0, stream>>>(fbonds, Wi, binput, msgA);

  float* src = msgA;
  float* dst = msgB;
  for (int d = 0; d < DEPTH - 1; ++d) {
    k_depth<<<NBONDS / 32, 256, 0, stream>>>(src, binput, bgraph, Wh, dst);
    float* t = src; src = dst; dst = t;
  }
  // latest message is in `src`
  k_atom<<<NATOMS / 32, 256, 0, stream>>>(src, fatoms, agraph, Wo, bo, atomh);
  k_pool<<<NMOLS, 128, 0, stream>>>(atomh, scope_start, scope_len, out);
}